// DMN_1666447311324
// MI455X (gfx1250) — compile-verified
//
#include <hip/hip_runtime.h>

#define NTREE 2048
#define BATCH 2048

typedef __attribute__((ext_vector_type(2))) float v2f;
typedef __attribute__((ext_vector_type(4))) float v4f;
typedef __attribute__((ext_vector_type(8))) float v8f;

struct Vec6 { float v[6]; };

// One-shot precompute: for each (node c, lane l) emit the per-lane WMMA A-operand quad.
// A(theta) is the 6x6 congruence map of R(theta) on packed symmetric 6-vectors
// (a = cos^2, b2 = sin^2, d = sqrt(2) cos sin). Per the 16x4 f32 A layout:
//   lanes 0-15  (row r): VGPRs {K0,K1} for WMMA#1, {K4,K5} for WMMA#2 -> (A[r][0],A[r][1],A[r][4],A[r][5])
//   lanes 16-31 (row r): VGPRs {K2,K3},            {K6,K7}=pad        -> (A[r][2],A[r][3],0,0)
__global__ void atab_kernel(const float* __restrict__ theta, v4f* __restrict__ tbl) {
  int t = blockIdx.x * blockDim.x + threadIdx.x;   // t = c*32 + l
  int c = t >> 5;
  int l = t & 31;
  if (c >= NTREE) return;
  float th = theta[c];
  float s = sinf(th);
  float co = cosf(th);
  float a = co * co, b2 = s * s, d = 1.41421356237309515f * co * s;
  float amb = a - b2, dd = d * d;
  int r = l & 15;
  float e0 = 0.f, e1 = 0.f, e2 = 0.f, e3 = 0.f, e4 = 0.f, e5 = 0.f;
  switch (r) {  // row r of A; rows >= 6 are zero
    case 0: e0 = a * a;   e1 = 2.f * a * b2;  e2 = -2.f * a * d;   e3 = b2 * b2; e4 = -2.f * b2 * d;  e5 = dd;        break;
    case 1: e0 = a * b2;  e1 = a * a + b2 * b2; e2 = d * amb;      e3 = a * b2;  e4 = -d * amb;       e5 = -dd;       break;
    case 2: e0 = a * d;   e1 = -d * amb;      e2 = a * amb - dd;   e3 = -b2 * d; e4 = b2 * amb + dd;  e5 = -d * amb;  break;
    case 3: e0 = b2 * b2; e1 = 2.f * a * b2;  e2 = 2.f * b2 * d;   e3 = a * a;   e4 = 2.f * a * d;    e5 = dd;        break;
    case 4: e0 = b2 * d;  e1 = d * amb;       e2 = b2 * amb + dd;  e3 = -a * d;  e4 = a * amb - dd;   e5 = d * amb;   break;
    case 5: e0 = dd;      e1 = -2.f * dd;     e2 = 2.f * d * amb;  e3 = dd;      e4 = -2.f * d * amb; e5 = amb * amb; break;
    default: break;
  }
  v4f q;
  if (l < 16) { q[0] = e0; q[1] = e1; q[2] = e4; q[3] = e5; }
  else        { q[0] = e2; q[1] = e3; q[2] = 0.f; q[3] = 0.f; }
  tbl[t] = q;
}

// Rotation as D(16x16) = A(16x4-padded) x B(4x16), V_WMMA_F32_16X16X4_F32, K split 4+2.
// A-operand comes straight from the precomputed table: one b128 load per lane.
// BCAST: rebroadcast result rows to lanes 16-31 (skip when result goes straight to a store).
template <bool BCAST>
__device__ __forceinline__ Vec6 wmma_rotate(const v4f* __restrict__ tbl, int c,
                                            const Vec6& in, int lane) {
  v4f q = tbl[c * 32 + lane];
  bool lo = lane < 16;
  v2f A1, A2, B1, B2;
  A1[0] = q[0]; A1[1] = q[1];
  A2[0] = q[2]; A2[1] = q[3];
  // B 4x16 f32 layout: lane&15 = column N; lanes 0-15 rows K0/K1, lanes 16-31 rows K2/K3.
  B1[0] = lo ? in.v[0] : in.v[2];  B1[1] = lo ? in.v[1] : in.v[3];
  B2[0] = lo ? in.v[4] : 0.f;      B2[1] = lo ? in.v[5] : 0.f;
  v8f acc = {};
  acc = __builtin_amdgcn_wmma_f32_16x16x4_f32(false, A1, false, B1, (short)0, acc, false, false);
  acc = __builtin_amdgcn_wmma_f32_16x16x4_f32(false, A2, false, B2, (short)0, acc, false, false);
  Vec6 out;
#pragma unroll
  for (int k = 0; k < 6; ++k) out.v[k] = BCAST ? __shfl(acc[k], lane & 15, 32) : acc[k];
  return out;
}

__device__ __forceinline__ Vec6 homogenise(const Vec6& D1, const Vec6& D2, float f1, float f2) {
  float gamma = f1 * D2.v[0] + f2 * D1.v[0];
  float g = (gamma == 0.f) ? 1.f : gamma;
  float ff = f1 * f2 / g;
  float u1 = D1.v[1] - D2.v[1];
  float u2 = D1.v[2] - D2.v[2];
  Vec6 o;
  o.v[0] = D1.v[0] * D2.v[0] / g;
  o.v[1] = (f1 * D1.v[1] * D2.v[0] + f2 * D2.v[1] * D1.v[0]) / g;
  o.v[2] = (f1 * D1.v[2] * D2.v[0] + f2 * D2.v[2] * D1.v[0]) / g;
  o.v[3] = f1 * D1.v[3] + f2 * D2.v[3] - ff * u1 * u1;
  o.v[4] = f1 * D1.v[4] + f2 * D2.v[4] - ff * u2 * u1;
  o.v[5] = f1 * D1.v[5] + f2 * D2.v[5] - ff * u2 * u2;
  return o;
}

// Bottom pass: one wave = (level-8 node, 16 batches). Computes 4 leaves, 2 level-9
// combines, 1 level-8 combine. Writes SoA buffer [6][256][BATCH].
__global__ void tree_bottom_kernel(const float* __restrict__ phase1,
                                   const float* __restrict__ phase2,
                                   const v4f* __restrict__ tbl,
                                   const float* __restrict__ frac,
                                   float* __restrict__ dst) {
  int lane = threadIdx.x & 31;
  int gwave = blockIdx.x * (blockDim.x >> 5) + (threadIdx.x >> 5);
  int ci = gwave >> 7;       // 0..255  (node c8 = 256 + ci)
  int tile = gwave & 127;    // 128 batch tiles of 16
  int b = tile * 16 + (lane & 15);

  Vec6 p1, p2;
#pragma unroll
  for (int k = 0; k < 6; ++k) { p1.v[k] = phase1[b * 6 + k]; p2.v[k] = phase2[b * 6 + k]; }

  int c8 = 256 + ci;
  int c9l = 2 * c8, c9r = 2 * c8 + 1;
  int x0 = 4 * c8;  // leaves x0..x0+3; even node -> phase1, odd -> phase2

  Vec6 L0 = wmma_rotate<true>(tbl, x0 + 0, p1, lane);
  Vec6 L1 = wmma_rotate<true>(tbl, x0 + 1, p2, lane);
  Vec6 L2 = wmma_rotate<true>(tbl, x0 + 2, p1, lane);
  Vec6 L3 = wmma_rotate<true>(tbl, x0 + 3, p2, lane);

  Vec6 h = homogenise(L0, L1, frac[NTREE - 2 * c9l], frac[NTREE - 2 * c9l - 1]);
  Vec6 Vl = wmma_rotate<true>(tbl, c9l, h, lane);
  h = homogenise(L2, L3, frac[NTREE - 2 * c9r], frac[NTREE - 2 * c9r - 1]);
  Vec6 Vr = wmma_rotate<true>(tbl, c9r, h, lane);
  h = homogenise(Vl, Vr, frac[NTREE - 2 * c8], frac[NTREE - 2 * c8 - 1]);
  Vec6 out = wmma_rotate<false>(tbl, c8, h, lane);  // store only reads lanes 0-15

  if (lane < 16) {
#pragma unroll
    for (int k = 0; k < 6; ++k) dst[(k * 256 + ci) * BATCH + b] = out.v[k];
  }
}

// Generic level pass: one wave = (node ci at this level, 16 batches).
// src: level L+1 SoA [6][2*numNodes][BATCH]; dst: [6][numNodes][BATCH], or AoS
// [BATCH][6] (d_out) when aos != 0.
__global__ void tree_level_kernel(const float* __restrict__ src, float* __restrict__ dst,
                                  int numNodes, int aos,
                                  const v4f* __restrict__ tbl,
                                  const float* __restrict__ frac) {
  int lane = threadIdx.x & 31;
  int gwave = blockIdx.x * (blockDim.x >> 5) + (threadIdx.x >> 5);
  int ci = gwave >> 7;
  int tile = gwave & 127;
  if (ci >= numNodes) return;
  int b = tile * 16 + (lane & 15);
  int nodes2 = numNodes * 2;

  Vec6 D1, D2;
#pragma unroll
  for (int k = 0; k < 6; ++k) {
    D1.v[k] = src[(k * nodes2 + 2 * ci)     * BATCH + b];
    D2.v[k] = src[(k * nodes2 + 2 * ci + 1) * BATCH + b];
  }
  int c = numNodes + ci;
  Vec6 h = homogenise(D1, D2, frac[NTREE - 2 * c], frac[NTREE - 2 * c - 1]);
  Vec6 out = wmma_rotate<false>(tbl, c, h, lane);

  if (lane < 16) {
    if (aos) {
#pragma unroll
      for (int k = 0; k < 6; ++k) dst[b * 6 + k] = out.v[k];
    } else {
#pragma unroll
      for (int k = 0; k < 6; ++k) dst[(k * numNodes + ci) * BATCH + b] = out.v[k];
    }
  }
}

extern "C" void kernel_launch(void* const* d_in, const int* in_sizes, int n_in,
                              void* d_out, int out_size, void* d_ws, size_t ws_size,
                              hipStream_t stream) {
  (void)in_sizes; (void)n_in; (void)out_size; (void)ws_size;
  const float* phase1 = (const float*)d_in[0];
  const float* phase2 = (const float*)d_in[1];
  const float* theta  = (const float*)d_in[2];
  const float* frac   = (const float*)d_in[5];
  float* ws = (float*)d_ws;

  // ws layout: [A-operand table: NTREE*32 float4 (1 MB)][level buffers L=8..1].
  v4f* tbl = (v4f*)ws;
  size_t cur = (size_t)NTREE * 32 * 4;   // floats
  size_t off[9];
  for (int L = 8; L >= 1; --L) { off[L] = cur; cur += (size_t)(1 << L) * BATCH * 6; }

  atab_kernel<<<(NTREE * 32) / 256, 256, 0, stream>>>(theta, tbl);

  {
    int waves = 256 * 128;
    int blocks = (waves * 32) / 256;  // 4096
    tree_bottom_kernel<<<blocks, 256, 0, stream>>>(phase1, phase2, tbl, frac, ws + off[8]);
  }
  for (int L = 7; L >= 1; --L) {
    int numNodes = 1 << L;
    int blocks = (numNodes * 128 * 32) / 256;  // numNodes * 16
    tree_level_kernel<<<blocks, 256, 0, stream>>>(ws + off[L + 1], ws + off[L],
                                                  numNodes, 0, tbl, frac);
  }
  {
    int blocks = (1 * 128 * 32) / 256;  // 16
    tree_level_kernel<<<blocks, 256, 0, stream>>>(ws + off[1], (float*)d_out,
                                                  1, 1, tbl, frac);
  }
}